// ReHashLoss_35442070126796
// MI455X (gfx1250) — compile-verified
//
#include <hip/hip_runtime.h>
#include <math.h>

// ---------------- problem constants (from reference) ----------------
#define NB        128        // batch rows
#define ND        500000     // database rows
#define KD        64         // code length
#define GAMMA_C   200.0f
#define TEMP_C    0.5f
#define ALPHA_C   12.0f
#define MARGIN_C  2.0f
#define LAM_C     1.0f
#define EPS_C     1e-12f

// ---------------- workspace layout (floats) ----------------
#define WS_U2    0     // [128] per-row |u|^2
#define WS_LP    128   // [128] sum of relu(d-(alpha-margin)) over positives
#define WS_CNT   256   // [128] positive count
#define WS_WSUM  384   // [128] sum of w over active negatives
#define WS_WAM   512   // [128] sum of w*(alpha-d)
#define WS_SQ    640   // [1]   sum of (uVt - 64 S)^2
#define WS_QUANT 641   // [1]   sum of (V_omega_u - u)^2
#define WS_INV   768   // [128] 1/(wsum+eps)

typedef __attribute__((ext_vector_type(2))) float v2f;
typedef __attribute__((ext_vector_type(8))) float v8f;

// ============ kernel 1: zero accumulators, |u_i|^2, quant sum ============
__global__ __launch_bounds__(256)
void rehash_prep(const float* __restrict__ u, const float* __restrict__ Vou,
                 float* ws) {
  int t = threadIdx.x;
  // zero indices [128, 641] (LP/CNT/WSUM/WAM/SQ/QUANT)
  for (int i = t; i < 514; i += 256) ws[128 + i] = 0.0f;
  if (t < NB) {
    const float* up = u + t * KD;
    float s = 0.0f;
#pragma unroll
    for (int k = 0; k < KD; ++k) s += up[k] * up[k];
    ws[WS_U2 + t] = s;
  }
  float q = 0.0f;
  for (int i = t; i < NB * KD; i += 256) {
    float d = Vou[i] - u[i];
    q += d * d;
  }
  __shared__ float red[256];
  red[t] = q;
  __syncthreads();
  for (int off = 128; off; off >>= 1) {
    if (t < off) red[t] += red[t + off];
    __syncthreads();
  }
  if (t == 0) ws[WS_QUANT] = red[0];
}

// ============ kernel 2: WMMA GEMM + fused epilogue ============
// Block = 256 threads = 8 wave32. Wave w covers rows [16w,16w+16) x 64 cols.
// A layout (V_WMMA_F32_16X16X4_F32, 16x4 f32): lane<16 -> row=lane, K {0,1};
// lane>=16 -> row=lane-16, K {2,3}. B (4x16) mirrored: lane<16 -> col=lane,
// K {0,1}; lane>=16 -> col=lane-16, K {2,3}.
// C/D layout: VGPR r: lanes 0-15 -> M=r, N=lane; lanes 16-31 -> M=r+8.
__global__ __launch_bounds__(256)
void rehash_main(const float* __restrict__ u, const float* __restrict__ V,
                 const float* __restrict__ S, const int* __restrict__ bind,
                 float* ws, float* __restrict__ Wout) {
  const int lane  = threadIdx.x & 31;
  const int wid   = threadIdx.x >> 5;
  const int half  = lane >> 4;
  const int l16   = lane & 15;
  const int rbase = wid * 16;
  const int colbase = blockIdx.x * 64;
  const int kofs  = half * 2;

  // Preload this wave's A stripe (u rows) into registers: 16 k-steps of K=4.
  const float* up = u + (rbase + l16) * KD + kofs;
  v2f areg[16];
#pragma unroll
  for (int ks = 0; ks < 16; ++ks) areg[ks] = *(const v2f*)(up + 4 * ks);

  float lp_s[8]  = {0,0,0,0,0,0,0,0};
  float cnt_s[8] = {0,0,0,0,0,0,0,0};
  float wsm_s[8] = {0,0,0,0,0,0,0,0};
  float wam_s[8] = {0,0,0,0,0,0,0,0};
  float sq_s = 0.0f;

  for (int nt = 0; nt < 4; ++nt) {
    int col16 = colbase + nt * 16;      // uniform across block
    if (col16 >= ND) break;             // ND % 16 == 0 -> tiles all-or-nothing
    int col = col16 + l16;
    const float* vp = V + (size_t)col * KD + kofs;

    v8f acc = {0.f,0.f,0.f,0.f,0.f,0.f,0.f,0.f};
    float bb = 0.0f;                    // on-the-fly |V[col]|^2 (this K half)
#pragma unroll
    for (int ks = 0; ks < 16; ++ks) {
      v2f b = *(const v2f*)(vp + 4 * ks);
      bb += b[0] * b[0] + b[1] * b[1];
      acc = __builtin_amdgcn_wmma_f32_16x16x4_f32(
          false, areg[ks], false, b, (short)0, acc, false, false);
    }
    float v2 = bb + __shfl_xor(bb, 16, 32);  // combine K halves (lane ^ 16)

    // Epilogue: acc[r] = uVt[row, col], row = rbase + r + 8*half
#pragma unroll
    for (int r = 0; r < 8; ++r) {
      int row    = rbase + r + 8 * half;
      float uv   = acc[r];
      float u2   = ws[WS_U2 + row];
      float s    = S[(size_t)row * ND + col];
      bool  self = (col == bind[row]);
      float d2   = u2 + v2 - 2.0f * uv;
      float dist = sqrtf(fmaxf(d2, 1e-12f));
      bool  pos  = (s > 0.0f) && !self;
      bool  neg  = (s <= 0.0f);
      float amd  = ALPHA_C - dist;
      bool  nact = neg && (dist < ALPHA_C);
      float w    = nact ? __expf(TEMP_C * amd) : 0.0f;
      lp_s[r]  += pos ? fmaxf(dist - (ALPHA_C - MARGIN_C), 0.0f) : 0.0f;
      cnt_s[r] += pos ? 1.0f : 0.0f;
      wsm_s[r] += w;
      wam_s[r] += w * amd;
      float dq = uv - (float)KD * s;
      sq_s += dq * dq;
      // exclusive cases: pos_alpha (+1) vs unnormalized -w (<= -1 when active)
      float outv = (pos && dist > (ALPHA_C - MARGIN_C)) ? 1.0f : -w;
      Wout[(size_t)row * ND + col] = outv;
    }
  }

  // Reduce per-row partials across the 16 lanes of each half, then atomics.
#pragma unroll
  for (int r = 0; r < 8; ++r) {
    float a0 = lp_s[r], a1 = cnt_s[r], a2 = wsm_s[r], a3 = wam_s[r];
#pragma unroll
    for (int off = 8; off; off >>= 1) {
      a0 += __shfl_xor(a0, off, 32);
      a1 += __shfl_xor(a1, off, 32);
      a2 += __shfl_xor(a2, off, 32);
      a3 += __shfl_xor(a3, off, 32);
    }
    if (l16 == 0) {
      int row = rbase + r + 8 * half;
      atomicAdd(&ws[WS_LP   + row], a0);
      atomicAdd(&ws[WS_CNT  + row], a1);
      atomicAdd(&ws[WS_WSUM + row], a2);
      atomicAdd(&ws[WS_WAM  + row], a3);
    }
  }

  // Block-level reduction of square-loss partial -> one atomic per block.
  __shared__ float red[8];
  float t = sq_s;
#pragma unroll
  for (int off = 16; off; off >>= 1) t += __shfl_xor(t, off, 32);
  if (lane == 0) red[wid] = t;
  __syncthreads();
  if (threadIdx.x == 0) {
    float s = 0.0f;
#pragma unroll
    for (int i = 0; i < 8; ++i) s += red[i];
    atomicAdd(&ws[WS_SQ], s);
  }
}

// ============ kernel 3: finalize scalar loss + per-row 1/(wsum+eps) ============
__global__ __launch_bounds__(128)
void rehash_finalize(float* ws, float* out) {
  __shared__ float red[128];
  int i = threadIdx.x;
  float wsum = ws[WS_WSUM + i];
  float Lp = ws[WS_LP + i] / fmaxf(ws[WS_CNT + i], 1.0f);
  float Ln = ws[WS_WAM + i] / (wsum + EPS_C);
  red[i] = Lp + LAM_C * Ln;
  ws[WS_INV + i] = 1.0f / (wsum + EPS_C);
  __syncthreads();
  for (int off = 64; off; off >>= 1) {
    if (i < off) red[i] += red[i + off];
    __syncthreads();
  }
  if (i == 0) {
    float loss_rh = red[0] / (float)NB;
    float sq = ws[WS_SQ] / ((float)NB * (float)ND);
    float qn = ws[WS_QUANT] / ((float)NB * (float)KD);
    out[0] = 1.0f * loss_rh + 1.0f * sq + GAMMA_C * qn;
  }
}

// ============ kernel 4: normalize negative entries of W ============
__global__ __launch_bounds__(256)
void rehash_normw(float* __restrict__ W, const float* __restrict__ ws) {
  int row = blockIdx.y;
  float inv = ws[WS_INV + row];
  float* p = W + (size_t)row * ND;
  for (int c = blockIdx.x * blockDim.x + threadIdx.x; c < ND;
       c += gridDim.x * blockDim.x) {
    float v = p[c];
    p[c] = (v < 0.0f) ? v * inv : v;
  }
}

// ================================ launch ================================
extern "C" void kernel_launch(void* const* d_in, const int* in_sizes, int n_in,
                              void* d_out, int out_size, void* d_ws,
                              size_t ws_size, hipStream_t stream) {
  const float* u    = (const float*)d_in[0];
  const float* V    = (const float*)d_in[1];
  const float* S    = (const float*)d_in[2];
  // d_in[3] = S_omega_u : unused by the math
  const float* Vou  = (const float*)d_in[4];
  const int*   bind = (const int*)d_in[5];
  float* out = (float*)d_out;
  float* ws  = (float*)d_ws;
  float* W   = out + 1;  // d_out = [loss, W(128 x 500000) row-major]

  rehash_prep<<<1, 256, 0, stream>>>(u, Vou, ws);
  int nblk = (ND + 63) / 64;  // 7813 column-tile blocks
  rehash_main<<<nblk, 256, 0, stream>>>(u, V, S, bind, ws, W);
  rehash_finalize<<<1, 128, 0, stream>>>(ws, out);
  rehash_normw<<<dim3(256, 128), 256, 0, stream>>>(W, ws);
}